// DeformByDepthConv2d_1305670058024
// MI455X (gfx1250) — compile-verified
//
#include <hip/hip_runtime.h>

typedef __attribute__((ext_vector_type(2)))  float    v2f;
typedef __attribute__((ext_vector_type(8)))  float    v8f;

#define BATCH 8
#define CIN   64
#define COUT  128
#define KSZ   9
#define KW_   3
#define H_    64
#define W_    64
#define HO    64
#define WO    64
#define NP    (HO*WO)      /* 4096 */
#define KKT   (CIN*KSZ)    /* 576  */

// ---------------------------------------------------------------------------
// Phase 1: deformable im2col (bilinear gather).
// One thread per (b, k, p); the 4 corner indices/weights are computed once and
// reused across all 64 input channels. Writes col[b][c*KSZ + k][p] (row-major
// [KKT][NP] per batch) so phase 2 reads B-matrix rows lane-contiguously.
// ---------------------------------------------------------------------------
__global__ __launch_bounds__(256) void deform_im2col(
    const float* __restrict__ rgb,       // [B, CIN, H, W]
    const float* __restrict__ offsets,   // [B, 2*KSZ, HO, WO]
    float* __restrict__ col)             // [B, KKT, NP]
{
  int tid = blockIdx.x * 256 + threadIdx.x;
  int p  = tid & (NP - 1);
  int bk = tid >> 12;                    // NP == 2^12
  if (bk >= BATCH * KSZ) return;
  int k = bk % KSZ;
  int b = bk / KSZ;
  int oy = p >> 6;                       // WO == 64
  int ox = p & 63;
  int ky = k / KW_;
  int kx = k % KW_;

  const float* offb = offsets + (size_t)b * (2 * KSZ) * NP;
  float dy = offb[(size_t)(2 * k)     * NP + p];
  float dx = offb[(size_t)(2 * k + 1) * NP + p];

  float y = (float)(oy - 1 + ky) + dy;   // STRIDE=1, PAD=1
  float x = (float)(ox - 1 + kx) + dx;
  float y0f = floorf(y);
  float x0f = floorf(x);
  float fy = y - y0f;
  float fx = x - x0f;
  int y0 = (int)y0f;
  int x0 = (int)x0f;
  int y1 = y0 + 1;
  int x1 = x0 + 1;

  float w00 = (1.f - fy) * (1.f - fx);
  float w01 = (1.f - fy) * fx;
  float w10 = fy * (1.f - fx);
  float w11 = fy * fx;

  bool vy0 = (y0 >= 0) && (y0 < H_);
  bool vy1 = (y1 >= 0) && (y1 < H_);
  bool vx0 = (x0 >= 0) && (x0 < W_);
  bool vx1 = (x1 >= 0) && (x1 < W_);
  if (!(vy0 && vx0)) w00 = 0.f;
  if (!(vy0 && vx1)) w01 = 0.f;
  if (!(vy1 && vx0)) w10 = 0.f;
  if (!(vy1 && vx1)) w11 = 0.f;

  int cy0 = min(max(y0, 0), H_ - 1);
  int cy1 = min(max(y1, 0), H_ - 1);
  int cx0 = min(max(x0, 0), W_ - 1);
  int cx1 = min(max(x1, 0), W_ - 1);
  int i00 = cy0 * W_ + cx0;
  int i01 = cy0 * W_ + cx1;
  int i10 = cy1 * W_ + cx0;
  int i11 = cy1 * W_ + cx1;

  const float* rb = rgb + (size_t)b * CIN * H_ * W_;
  float* cb = col + (size_t)b * KKT * NP + (size_t)k * NP + p;
  #pragma unroll 4
  for (int c = 0; c < CIN; ++c) {
    const float* pl = rb + (size_t)c * (H_ * W_);
    float v = w00 * pl[i00] + w01 * pl[i01] + w10 * pl[i10] + w11 * pl[i11];
    cb[(size_t)c * (KSZ * NP)] = v;
  }
}

// ---------------------------------------------------------------------------
// Phase 2: out[b] = W[128x576] x col_b[576x4096] + bias, in full fp32 via
// V_WMMA_F32_16X16X4_F32 (wave32). Each wave owns a 32(M) x 64(N) tile:
// 2 M-tiles x 4 N-tiles = 8 accumulators. Per K-step of 4: two b64 A-fragment
// loads + eight b32 B-fragment loads feed 8 WMMAs (~1.25 VMEM per WMMA),
// with A reused 4x and B reused 2x in registers.
//
// Fragment layouts (ISA 7.12.2):
//   A 16x4:  lane ll holds row M=ll, VGPR pair = K{2lh, 2lh+1}
//   B 4x16:  VGPR pair = rows K{2lh, 2lh+1}, col N = ll
//   C 16x16: VGPR i -> M = i + 8*lh, N = ll
// ---------------------------------------------------------------------------
__global__ __launch_bounds__(256) void gemm_wmma(
    const float* __restrict__ wmat,   // [COUT, KKT]
    const float* __restrict__ col,    // [B, KKT, NP]
    const float* __restrict__ bias,   // [COUT]
    float* __restrict__ out)          // [B, COUT, NP]
{
  const int wave = (blockIdx.x * 256 + threadIdx.x) >> 5;
  const int lane = threadIdx.x & 31;
  const int ll = lane & 15;
  const int lh = lane >> 4;

  int nstrip = wave & 63;            // 64 strips of 64 columns
  int mtile  = (wave >> 6) & 3;      // 4 tiles of 32 rows
  int b      = wave >> 8;
  if (b >= BATCH) return;

  const int m0 = mtile * 32;
  const int n0 = nstrip * 64;
  const float* A0 = wmat + (size_t)(m0 + ll) * KKT;       // rows m0+ll
  const float* A1 = A0 + (size_t)16 * KKT;                // rows m0+16+ll
  const float* Bm = col + (size_t)b * KKT * NP + n0;
  float* Cp = out + (size_t)b * COUT * NP + (size_t)m0 * NP + n0;

  v8f acc00 = {}, acc01 = {}, acc02 = {}, acc03 = {};
  v8f acc10 = {}, acc11 = {}, acc12 = {}, acc13 = {};

  for (int kk = 0; kk < KKT; kk += 4) {
    v2f a0 = *(const v2f*)(A0 + kk + 2 * lh);             // K = {2lh, 2lh+1}
    v2f a1 = *(const v2f*)(A1 + kk + 2 * lh);
    const float* Bk0 = Bm + (size_t)(kk + 2 * lh) * NP + ll;
    const float* Bk1 = Bk0 + NP;
    v2f b0, b1, b2, b3;
    b0.x = Bk0[0];  b0.y = Bk1[0];
    b1.x = Bk0[16]; b1.y = Bk1[16];
    b2.x = Bk0[32]; b2.y = Bk1[32];
    b3.x = Bk0[48]; b3.y = Bk1[48];
    acc00 = __builtin_amdgcn_wmma_f32_16x16x4_f32(false, a0, false, b0, (short)0, acc00, false, false);
    acc01 = __builtin_amdgcn_wmma_f32_16x16x4_f32(false, a0, false, b1, (short)0, acc01, false, false);
    acc02 = __builtin_amdgcn_wmma_f32_16x16x4_f32(false, a0, false, b2, (short)0, acc02, false, false);
    acc03 = __builtin_amdgcn_wmma_f32_16x16x4_f32(false, a0, false, b3, (short)0, acc03, false, false);
    acc10 = __builtin_amdgcn_wmma_f32_16x16x4_f32(false, a1, false, b0, (short)0, acc10, false, false);
    acc11 = __builtin_amdgcn_wmma_f32_16x16x4_f32(false, a1, false, b1, (short)0, acc11, false, false);
    acc12 = __builtin_amdgcn_wmma_f32_16x16x4_f32(false, a1, false, b2, (short)0, acc12, false, false);
    acc13 = __builtin_amdgcn_wmma_f32_16x16x4_f32(false, a1, false, b3, (short)0, acc13, false, false);
  }

  // Epilogue: C/D layout VGPR i -> M = i + 8*lh, N = ll.
  const float* bs = bias + m0;
  #pragma unroll
  for (int i = 0; i < 8; ++i) {
    int m = i + 8 * lh;
    float bv0 = bs[m];
    float bv1 = bs[m + 16];
    float* r0 = Cp + (size_t)m * NP + ll;
    float* r1 = r0 + (size_t)16 * NP;
    r0[0]  = acc00[i] + bv0;
    r0[16] = acc01[i] + bv0;
    r0[32] = acc02[i] + bv0;
    r0[48] = acc03[i] + bv0;
    r1[0]  = acc10[i] + bv1;
    r1[16] = acc11[i] + bv1;
    r1[32] = acc12[i] + bv1;
    r1[48] = acc13[i] + bv1;
  }
}

extern "C" void kernel_launch(void* const* d_in, const int* in_sizes, int n_in,
                              void* d_out, int out_size, void* d_ws, size_t ws_size,
                              hipStream_t stream) {
  const float* rgb     = (const float*)d_in[0];  // [8,64,64,64]
  const float* offsets = (const float*)d_in[1];  // [8,18,64,64]
  const float* weight  = (const float*)d_in[2];  // [128,64,3,3]
  const float* bias    = (const float*)d_in[3];  // [128]
  float* out = (float*)d_out;                    // [8,128,64,64]
  float* col = (float*)d_ws;                     // [8,576,4096] f32 = 72 MB

  (void)in_sizes; (void)n_in; (void)out_size; (void)ws_size;

  {
    int total  = BATCH * KSZ * NP;                // 294912 threads
    int blocks = (total + 255) / 256;
    deform_im2col<<<blocks, 256, 0, stream>>>(rgb, offsets, col);
  }
  {
    int waves  = BATCH * (COUT / 32) * (NP / 64); // 2048 waves
    int blocks = waves / 8;                       // 8 waves per 256-thread block
    gemm_wmma<<<blocks, 256, 0, stream>>>(weight, col, bias, out);
  }
}